// Model_21139829031540
// MI455X (gfx1250) — compile-verified
//
#include <hip/hip_runtime.h>
#include <hip/hip_bf16.h>
#include <stdint.h>

#define N_TOKEN 1024
#define C_Z     128
#define IN_DIM  139
#define R_MAX   32
#define S_MAX   2

typedef __attribute__((ext_vector_type(4))) float        v4f;
typedef __attribute__((ext_vector_type(4))) unsigned int v4u;
typedef __attribute__((ext_vector_type(8))) int          v8i;
typedef __attribute__((ext_vector_type(4))) int          v4i;

#define SW_BYTES   (IN_DIM * C_Z * 4)                  /* 71168: transposed W  */
#define STG_BYTES  (IN_DIM * C_Z * 4)                  /* 71168: raw W staging */
#define IDX_BYTES  (5 * N_TOKEN * 4)                   /* 20480: index arrays  */
#define SMEM_BYTES (SW_BYTES + STG_BYTES + IDX_BYTES)  /* 162816 < 320KB WGP   */

#if defined(__has_builtin)
#  if __has_builtin(__builtin_amdgcn_tensor_load_to_lds)
#    define HAVE_TDM 1
#  endif
#  if __has_builtin(__builtin_amdgcn_s_wait_tensorcnt)
#    define HAVE_TCNT 1
#  endif
#endif

__device__ __forceinline__ uint32_t lds_off_u32(const void* p) {
  // LDS aperture: generic addr[31:0] is the LDS byte offset.
  return (uint32_t)(uintptr_t)p;
}

__global__ __launch_bounds__(256) void relpos_kernel(
    const int* __restrict__ asym, const int* __restrict__ resi,
    const int* __restrict__ ent,  const int* __restrict__ tok,
    const int* __restrict__ sym,  const float* __restrict__ W,
    float* __restrict__ out)
{
  extern __shared__ char smem[];
  float* sW     = (float*)smem;                          // [IN_DIM][C_Z] (transposed)
  float* sStage = (float*)(smem + SW_BYTES);             // [C_Z][IN_DIM] (raw)
  int*   sIdx   = (int*)  (smem + SW_BYTES + STG_BYTES); // 5 x N_TOKEN

  const int tid = threadIdx.x;
  const int i   = blockIdx.x;

  // ---- ASYNCcnt path: async-copy the 5 index arrays (4KB each) into LDS ----
  // 256 threads x 16B = 4KB per array; one b128 async per thread per array.
  {
    const uint32_t base = lds_off_u32(sIdx);
    const uint32_t voff = (uint32_t)tid * 16u;
    const uint64_t ga[5] = { (uint64_t)(uintptr_t)asym, (uint64_t)(uintptr_t)resi,
                             (uint64_t)(uintptr_t)ent,  (uint64_t)(uintptr_t)tok,
                             (uint64_t)(uintptr_t)sym };
    #pragma unroll
    for (int a = 0; a < 5; ++a) {
      uint32_t ldsa = base + (uint32_t)(a * N_TOKEN * 4) + voff;
      asm volatile("global_load_async_to_lds_b128 %0, %1, %2"
                   :: "v"(ldsa), "v"(voff), "s"(ga[a]) : "memory");
    }
  }

  // ---- TENSORcnt path: TDM DMA of W (row-major [128][139] f32) into LDS ----
#if defined(HAVE_TDM)
  if (tid < 32) {  // one wave issues the DMA; EXEC-independent, SGPR descriptor
    union { unsigned long long q[2]; v4u v; } g0;
    union { unsigned long long q[4]; v8i v; } g1;
    union { unsigned long long q[2]; v4i v; } gz4;
    union { unsigned long long q[4]; v8i v; } gz8;
    const uint64_t ga = (uint64_t)(uintptr_t)W;
    // Group0: count=1 | lds_addr[63:32] ; global_addr[120:64] | type=2 [127:126]
    g0.q[0] = 1ull | ((uint64_t)lds_off_u32(sStage) << 32);
    g0.q[1] = (ga & ((1ull << 57) - 1)) | (2ull << 62);
    // Group1: data_size=2(4B) [17:16]; tensor_dim0 [79:48]; tensor_dim1 [111:80];
    //         tile_dim0 [127:112]; tile_dim1 [143:128]; dim0_stride [207:160]
    g1.q[0] = (2ull << 16) | ((uint64_t)(IN_DIM & 0xFFFF) << 48);
    g1.q[1] = ((uint64_t)IN_DIM >> 16)
            | ((uint64_t)C_Z   << 16)
            | ((uint64_t)IN_DIM << 48);
    g1.q[2] = (uint64_t)C_Z | ((uint64_t)IN_DIM << 32);
    g1.q[3] = 0;
    gz4.q[0] = 0; gz4.q[1] = 0;
    gz8.q[0] = 0; gz8.q[1] = 0; gz8.q[2] = 0; gz8.q[3] = 0;
    // 6-arg form (clang-23 / therock-10.0): (g0, g1, g2, g3, g4, cpol)
    __builtin_amdgcn_tensor_load_to_lds(g0.v, g1.v, gz4.v, gz4.v, gz8.v, 0);
#  if defined(HAVE_TCNT)
    __builtin_amdgcn_s_wait_tensorcnt(0);
#  else
    asm volatile("s_wait_tensorcnt 0" ::: "memory");
#  endif
  }
#else
  for (int f = tid; f < IN_DIM * C_Z; f += 256) sStage[f] = W[f];
#endif

  asm volatile("s_wait_asynccnt 0" ::: "memory");
  __syncthreads();

  // ---- transpose staging [c][d] -> sW [d][c] (row gathers become b128) ----
  for (int f = tid; f < IN_DIM * C_Z; f += 256) {
    int c = f / IN_DIM;
    int d = f - c * IN_DIM;
    sW[d * C_Z + c] = sStage[f];
  }
  __syncthreads();

  // ---- per-i scalars (uniform -> s_load) ----
  const int ai = asym[i], ri = resi[i], ei = ent[i], ti = tok[i], si = sym[i];

  const int wave = tid >> 5;
  const int lane = tid & 31;
  const v4f* sW4 = (const v4f*)sW;              // rows of 32 x float4 (512B)
  const v4f  rEnt = sW4[132 * 32 + lane];       // j-independent entity row

  const int* sAsym = sIdx;
  const int* sRes  = sIdx + 1 * N_TOKEN;
  const int* sEnt  = sIdx + 2 * N_TOKEN;
  const int* sTok  = sIdx + 3 * N_TOKEN;
  const int* sSym  = sIdx + 4 * N_TOKEN;

  const int j0 = wave * (N_TOKEN / 8);
  float* outBase = out + (size_t)i * N_TOKEN * C_Z + (size_t)lane * 4;

  #pragma unroll 2
  for (int j = j0; j < j0 + N_TOKEN / 8; ++j) {
    const int aj = sAsym[j], rj = sRes[j], ej = sEnt[j], tj = sTok[j], sj = sSym[j];
    const bool sc = (ai == aj);
    const bool sr = (ri == rj);
    const bool se = (ei == ej);

    int dres = min(max(ri - rj + R_MAX, 0), 2 * R_MAX);
    if (!sc) dres = 2 * R_MAX + 1;                       // [0,65]
    int dtok = min(max(ti - tj + R_MAX, 0), 2 * R_MAX);
    if (!(sc && sr)) dtok = 2 * R_MAX + 1;               // [0,65]
    int dch = min(max(si - sj + S_MAX, 0), 2 * S_MAX);
    if (!se) dch = 2 * S_MAX + 1;                        // [0,5]

    const v4f a = sW4[dres * 32 + lane];                 // WT[d_res]
    const v4f b = sW4[(66 + dtok) * 32 + lane];          // WT[66+d_tok]
    const v4f d = sW4[(133 + dch) * 32 + lane];          // WT[133+d_ch]
    const float sef = se ? 1.0f : 0.0f;

    v4f v = a + b + d + sef * rEnt;

    // write-once 512MB output: non-temporal b128 stores
    __builtin_nontemporal_store(v, (v4f*)(outBase + (size_t)j * C_Z));
  }
}

extern "C" void kernel_launch(void* const* d_in, const int* in_sizes, int n_in,
                              void* d_out, int out_size, void* d_ws, size_t ws_size,
                              hipStream_t stream) {
  (void)in_sizes; (void)n_in; (void)d_ws; (void)ws_size; (void)out_size;
  const int*   asym = (const int*)d_in[0];
  const int*   resi = (const int*)d_in[1];
  const int*   ent  = (const int*)d_in[2];
  const int*   tok  = (const int*)d_in[3];
  const int*   sym  = (const int*)d_in[4];
  const float* W    = (const float*)d_in[5];
  float*       out  = (float*)d_out;

  // Dynamic LDS above the 64KB static cap (CDNA5: up to 320KB per workgroup).
  (void)hipFuncSetAttribute(reinterpret_cast<const void*>(relpos_kernel),
                            hipFuncAttributeMaxDynamicSharedMemorySize, SMEM_BYTES);

  relpos_kernel<<<dim3(N_TOKEN), dim3(256), SMEM_BYTES, stream>>>(
      asym, resi, ent, tok, sym, W, out);
}